// SegNetNorm_8839042695321
// MI455X (gfx1250) — compile-verified
//
#include <hip/hip_runtime.h>
#include <math.h>
#include <stdint.h>

typedef __attribute__((ext_vector_type(2))) float v2f;
typedef __attribute__((ext_vector_type(8))) float v8f;

// Problem constants (from setup_inputs)
constexpr int kB  = 2;
constexpr int kNs = 8192;
constexpr int kNt = 2048;
constexpr int kK  = 64;   // neighbors
constexpr int kC  = 64;   // channels
constexpr int kP  = 3;    // prop neighbors
constexpr int kYS = 48;   // 16 SH comps * 3 shells

// ---------------------------------------------------------------------------
// Kernel 1: per-(b,n) SH-Gaussian kernel construction + K-contraction matmul
//   y[b,n,ys,c] = (1/K) * sum_k  kernel[b,n,k,ys] * feats[b, idx[b,n,k], c]
// 48x64 output = A^T(48xK) x B(K=64 x C=64) via v_wmma_f32_16x16x4_f32.
// Neighbor-feature gather uses CDNA5 async global->LDS DMA (ASYNCcnt).
// ---------------------------------------------------------------------------
__global__ __launch_bounds__(128) void conv_wmma_kernel(
    const float* __restrict__ src_feats,   // (B,Ns,64)
    const float* __restrict__ patches,     // (B,Nt,64,3)
    const float* __restrict__ pdist,       // (B,Nt,64)
    const int*   __restrict__ pidx,        // (B,Nt,64)
    float*       __restrict__ y_ws)        // (B,Nt,48,64)
{
    __shared__ float As[kK][kYS];          // [k][ys]  12 KB
    __shared__ float Bs[kK][kC];           // [k][c]   16 KB

    const int bn  = blockIdx.x;            // 0 .. B*Nt-1
    const int tid = threadIdx.x;

    // ---- Phase 1b (issue first): async gather of neighbor features --------
    // Each thread DMAs 8 x 16B chunks of one half-row straight into LDS via
    // GLOBAL_LOAD_ASYNC_TO_LDS_B128: no data VGPRs, overlaps with the SH
    // computation below. IOFFSET is added to both LDS and global addresses.
    {
        const int row  = tid >> 1;         // 0..63 (k)
        const int half = tid & 1;          // 32-float half of the channel row
        const int idx  = pidx[(size_t)bn * kK + row];
        const int b    = bn / kNt;
        const float* fp =
            src_feats + ((size_t)b * kNs + idx) * kC + half * 32;
        const uint64_t gaddr    = (uint64_t)(uintptr_t)fp;
        const uint32_t lds_base = (uint32_t)(uintptr_t)&Bs[row][half * 32];
        #pragma unroll
        for (int i = 0; i < 8; ++i) {
            asm volatile(
                "global_load_async_to_lds_b128 %0, %1, off offset:%2"
                :
                : "v"(lds_base), "v"(gaddr), "i"(i * 16)
                : "memory");
        }
    }

    // ---- Phase 1a: SH * gaussian-shell kernel rows (one k per thread) ----
    if (tid < kK) {
        const int k = tid;
        const size_t base = (size_t)bn * kK + k;
        const float px = patches[base * 3 + 0];
        const float py = patches[base * 3 + 1];
        const float pz = patches[base * 3 + 2];
        const float d  = pdist[base];
        const float inv = 1.0f / (d + 1e-8f);
        const float x = px * inv, y = py * inv, z = pz * inv;
        const float x2 = x * x, y2 = y * y, z2 = z * z;

        float sh[16];
        sh[0]  = 0.28209479177387814f;
        sh[1]  = 0.4886025119029199f * y;
        sh[2]  = 0.4886025119029199f * z;
        sh[3]  = 0.4886025119029199f * x;
        sh[4]  = 1.0925484305920792f * x * y;
        sh[5]  = 1.0925484305920792f * y * z;
        sh[6]  = 0.31539156525252005f * (3.0f * z2 - 1.0f);
        sh[7]  = 1.0925484305920792f * x * z;
        sh[8]  = 0.5462742152960396f * (x2 - y2);
        sh[9]  = 0.5900435899266435f * y * (3.0f * x2 - y2);
        sh[10] = 2.890611442640554f * x * y * z;
        sh[11] = 0.4570457994644658f * y * (5.0f * z2 - 1.0f);
        sh[12] = 0.3731763325901154f * z * (5.0f * z2 - 3.0f);
        sh[13] = 0.4570457994644658f * x * (5.0f * z2 - 1.0f);
        sh[14] = 1.445305721320277f * z * (x2 - y2);
        sh[15] = 0.5900435899266435f * x * (x2 - 3.0f * y2);

        // gaussian shell weights, shells at 0, 0.5, 1.0
        const float GS = 6.23832462504f;   // ln(2) * NUM_SHELLS^2
        float w[3];
        float wsum = 0.0f;
        #pragma unroll
        for (int s = 0; s < 3; ++s) {
            const float dd = d - 0.5f * (float)s;
            w[s] = __expf(-GS * dd * dd);
            wsum += w[s];
        }
        const float mask = (d <= 1.0f) ? 1.0f : 0.0f;
        const float norm = mask / (wsum + 1e-8f);
        #pragma unroll
        for (int s = 0; s < 3; ++s) w[s] *= norm;

        #pragma unroll
        for (int yy = 0; yy < 16; ++yy)
            #pragma unroll
            for (int s = 0; s < 3; ++s)
                As[k][yy * 3 + s] = sh[yy] * w[s];
    }

    // Drain this wave's async DMA, then synchronize the workgroup so all
    // waves see both As (ds writes) and Bs (async DMA writes).
    asm volatile("s_wait_asynccnt 0x0" ::: "memory");
    __syncthreads();

    // ---- Phase 2: 48x64 = A^T(48xK) x B(KxC) via WMMA f32 16x16x4 ----------
    // wave w owns column tile [16w, 16w+16); loops 3 row tiles of ys.
    const int lane = tid & 31;
    const int wave = tid >> 5;                 // 0..3
    const int col  = wave * 16 + (lane & 15);  // N index in C
    const int koff = (lane < 16) ? 0 : 2;      // A/B fragment K sub-offset
    const int ml   = lane & 15;

    #pragma unroll
    for (int mt = 0; mt < 3; ++mt) {
        v8f acc = {};
        const int arow = mt * 16 + ml;         // ys row for this lane
        #pragma unroll
        for (int k = 0; k < kK; k += 4) {
            v2f a, b;
            // A 16x4 f32 layout: lanes0-15 hold K=0,1; lanes16-31 hold K=2,3.
            a.x = As[k + koff    ][arow];
            a.y = As[k + koff + 1][arow];
            // B 4x16 f32 layout mirrors A with N striped across lanes.
            b.x = Bs[k + koff    ][col];
            b.y = Bs[k + koff + 1][col];
            acc = __builtin_amdgcn_wmma_f32_16x16x4_f32(
                false, a, false, b, (short)0, acc, false, false);
        }
        // D layout: VGPR j -> M = j + (lane<16 ? 0 : 8), N = lane&15
        const int   mbase = (lane < 16) ? 0 : 8;
        const float scale = 1.0f / (float)kK;
        const size_t obase =
            ((size_t)bn * kYS + (size_t)(mt * 16 + mbase)) * kC + col;
        #pragma unroll
        for (int j = 0; j < 8; ++j)
            y_ws[obase + (size_t)j * kC] = acc[j] * scale;
    }
}

// ---------------------------------------------------------------------------
// Kernel 2: feature propagation (inverse-square-distance blend over P=3
// targets) fused with per-degree L2 norms over the m axis.
// One thread per (b, s, ch) with ch = shell*64 + c  (192 channels).
// ---------------------------------------------------------------------------
__global__ __launch_bounds__(256) void prop_norm_kernel(
    const float* __restrict__ y_ws,        // (B,Nt,48,64)
    const int*   __restrict__ prop_idx,    // (B,Ns,3)
    const float* __restrict__ prop_dist,   // (B,Ns,3)
    float*       __restrict__ out)         // (B,Ns,768)
{
    const int total = kB * kNs * 192;
    const int g = blockIdx.x * 256 + threadIdx.x;
    if (g >= total) return;

    const int ch = g % 192;
    const int bs = g / 192;                // b*Ns + s
    const int b  = bs / kNs;
    const int shell = ch >> 6;             // ch / 64
    const int c     = ch & 63;

    const size_t pb = (size_t)bs * kP;
    float w[kP];
    float wsum = 0.0f;
    const float* ptr[kP];
    #pragma unroll
    for (int p = 0; p < kP; ++p) {
        const float d = prop_dist[pb + p];
        w[p] = 1.0f / (d * d + 1e-8f);
        wsum += w[p];
        const int t = prop_idx[pb + p];
        // y viewed as (b, n, m16, shell*64 + c):  ys = m16*3 + shell
        ptr[p] = y_ws + ((size_t)(b * kNt + t) * kYS + shell) * kC + c;
    }
    const float invs = 1.0f / wsum;
    #pragma unroll
    for (int p = 0; p < kP; ++p) w[p] *= invs;

    float dacc[4] = {0.0f, 0.0f, 0.0f, 0.0f};
    #pragma unroll
    for (int m = 0; m < 16; ++m) {
        float v = 0.0f;
        #pragma unroll
        for (int p = 0; p < kP; ++p) v += w[p] * ptr[p][(size_t)m * 192];
        const int l = (m == 0) ? 0 : (m < 4) ? 1 : (m < 9) ? 2 : 3;
        dacc[l] += v * v;
    }

    float* ob = out + (size_t)bs * 768 + ch;
    #pragma unroll
    for (int l = 0; l < 4; ++l)
        ob[(size_t)l * 192] = sqrtf(fmaxf(dacc[l], 1e-8f));
}

extern "C" void kernel_launch(void* const* d_in, const int* in_sizes, int n_in,
                              void* d_out, int out_size, void* d_ws, size_t ws_size,
                              hipStream_t stream) {
    const float* src_feats = (const float*)d_in[0];  // (B,Ns,64)
    const float* patches   = (const float*)d_in[1];  // (B,Nt,64,3)
    const float* pdist     = (const float*)d_in[2];  // (B,Nt,64)
    const int*   pidx      = (const int*)  d_in[3];  // (B,Nt,64)
    const int*   prop_idx  = (const int*)  d_in[4];  // (B,Ns,3)
    const float* prop_dist = (const float*)d_in[5];  // (B,Ns,3)

    float* y_ws = (float*)d_ws;                      // (B,Nt,48,64) = 50.3 MB

    conv_wmma_kernel<<<kB * kNt, 128, 0, stream>>>(
        src_feats, patches, pdist, pidx, y_ws);

    const int total = kB * kNs * 192;
    prop_norm_kernel<<<(total + 255) / 256, 256, 0, stream>>>(
        y_ws, prop_idx, prop_dist, (float*)d_out);
}